// MultiF1Score_9105330668074
// MI455X (gfx1250) — compile-verified
//
#include <hip/hip_runtime.h>

// MultiF1Score for MI455X (gfx1250, wave32).
// Memory-bound: 404 MB streamed once -> ~17us floor at 23.3 TB/s.
// Pass 1: wave-per-row argmax (25 lanes x float4 = 100 cols), LDS histograms.
// Pass 2: 100-class F1 + mean via V_WMMA_F32_16X16X4_F32 row-sum reduction.

#define NUM_C 100
#define EPSF 1e-7f

typedef __attribute__((ext_vector_type(2))) float v2f;
typedef __attribute__((ext_vector_type(8))) float v8f;

__global__ void mf1_zero(unsigned* cnt) {
  int t = threadIdx.x;
  if (t < 384) cnt[t] = 0u;
}

__device__ __forceinline__ int imin_(int a, int b) { return a < b ? a : b; }

// One wave handles one row: lanes 0..24 hold cols 4L..4L+3 in a float4,
// lanes 25..31 hold -inf.  Two-pass argmax: shuffle-max, then ballot+ffs
// (SALU) for the first-matching column (== jnp.argmax tie-break).
__device__ __forceinline__ void mf1_process_row(
    float4 v, long long row, int lane, const int* labels,
    unsigned* s_pred, unsigned* s_true, unsigned* s_tp) {
  float mx = fmaxf(fmaxf(v.x, v.y), fmaxf(v.z, v.w));
#pragma unroll
  for (int off = 16; off >= 1; off >>= 1)
    mx = fmaxf(mx, __shfl_xor(mx, off, 32));
  unsigned b0 = (unsigned)__ballot(v.x == mx);
  unsigned b1 = (unsigned)__ballot(v.y == mx);
  unsigned b2 = (unsigned)__ballot(v.z == mx);
  unsigned b3 = (unsigned)__ballot(v.w == mx);
  int best = 0x7fffffff;
  if (b0) best = imin_(best, 4 * (__ffs(b0) - 1) + 0);
  if (b1) best = imin_(best, 4 * (__ffs(b1) - 1) + 1);
  if (b2) best = imin_(best, 4 * (__ffs(b2) - 1) + 2);
  if (b3) best = imin_(best, 4 * (__ffs(b3) - 1) + 3);
  if (lane == 0) {
    int lab = labels[row];
    atomicAdd(&s_pred[best], 1u);
    atomicAdd(&s_true[lab], 1u);
    if (best == lab) atomicAdd(&s_tp[lab], 1u);
  }
}

__global__ __launch_bounds__(256) void mf1_main(
    const float* __restrict__ logits, const int* __restrict__ labels,
    unsigned* __restrict__ g_pred, unsigned* __restrict__ g_true,
    unsigned* __restrict__ g_tp, int N) {
  __shared__ unsigned s_pred[NUM_C], s_true[NUM_C], s_tp[NUM_C];
  for (int i = threadIdx.x; i < NUM_C; i += blockDim.x) {
    s_pred[i] = 0u; s_true[i] = 0u; s_tp[i] = 0u;
  }
  __syncthreads();

  const int lane = threadIdx.x & 31;
  const long long wave =
      (long long)blockIdx.x * (blockDim.x >> 5) + (threadIdx.x >> 5);
  const long long nwaves = (long long)gridDim.x * (blockDim.x >> 5);
  const float NEG = -__builtin_inff();

  // 2-row unroll: two b128 loads in flight per wave-iteration.
  for (long long row = wave * 2; row < N; row += nwaves * 2) {
    float4 v0 = make_float4(NEG, NEG, NEG, NEG);
    float4 v1 = v0;
    const bool has1 = (row + 1) < N;
    if (lane < 25) {
      v0 = ((const float4*)(logits + (size_t)row * NUM_C))[lane];
      if (has1)
        v1 = ((const float4*)(logits + (size_t)(row + 1) * NUM_C))[lane];
    }
    long long pf = row + nwaves * 2;
    if (pf < N)  // global_prefetch_b8 for the next strided pair
      __builtin_prefetch(logits + (size_t)pf * NUM_C + lane * 4, 0, 1);

    mf1_process_row(v0, row, lane, labels, s_pred, s_true, s_tp);
    if (has1) mf1_process_row(v1, row + 1, lane, labels, s_pred, s_true, s_tp);
  }
  __syncthreads();
  for (int i = threadIdx.x; i < NUM_C; i += blockDim.x) {
    if (s_pred[i]) atomicAdd(&g_pred[i], s_pred[i]);
    if (s_true[i]) atomicAdd(&g_true[i], s_true[i]);
    if (s_tp[i])   atomicAdd(&g_tp[i],   s_tp[i]);
  }
}

__device__ __forceinline__ float mf1_f1(int c, const unsigned* pc,
                                        const unsigned* tc, const unsigned* tp) {
  float tpv = (float)tp[c];
  float fpv = (float)pc[c] - tpv;
  float fnv = (float)tc[c] - tpv;
  float p = tpv / (tpv + fpv + EPSF);
  float r = tpv / (tpv + fnv + EPSF);
  return 2.0f * p * r / (p + r + EPSF);
}

// One wave.  Pack the 100 f1 values (zero-padded to 128) into two 16x4 f32
// A-matrices; D = A x Ones(4x16) + C broadcasts row-sums, chained over both
// tiles.  Per-lane sum of D's 8 VGPRs gives half-sums; one xor-16 shuffle
// completes the total.
__global__ void mf1_final(const unsigned* __restrict__ pc,
                          const unsigned* __restrict__ tc,
                          const unsigned* __restrict__ tp,
                          float* __restrict__ out) {
  const int lane = threadIdx.x;          // 0..31
  const int m = lane & 15;               // A-matrix row
  const int kb = (lane >> 4) << 1;       // K base: lanes 0-15 -> K0,K1; 16-31 -> K2,K3
  v2f a1, a2;
#pragma unroll
  for (int j = 0; j < 2; ++j) {
    int k = kb + j;
    int c1 = m + 16 * k;        // 0..63
    int c2 = 64 + m + 16 * k;   // 64..127
    a1[j] = mf1_f1(c1, pc, tc, tp);
    a2[j] = (c2 < NUM_C) ? mf1_f1(c2, pc, tc, tp) : 0.0f;
  }
  v2f ones; ones[0] = 1.0f; ones[1] = 1.0f;
  v8f acc = {};
  acc = __builtin_amdgcn_wmma_f32_16x16x4_f32(
      false, a1, false, ones, (short)0, acc, false, false);
  acc = __builtin_amdgcn_wmma_f32_16x16x4_f32(
      false, a2, false, ones, (short)0, acc, false, false);
  float t = acc[0] + acc[1] + acc[2] + acc[3] +
            acc[4] + acc[5] + acc[6] + acc[7];
  t += __shfl_xor(t, 16, 32);
  if (lane == 0) out[0] = t * (1.0f / (float)NUM_C);
}

extern "C" void kernel_launch(void* const* d_in, const int* in_sizes, int n_in,
                              void* d_out, int out_size, void* d_ws, size_t ws_size,
                              hipStream_t stream) {
  const float* logits = (const float*)d_in[0];
  const int*   labels = (const int*)d_in[1];
  const int N = in_sizes[1];  // 1,000,000

  unsigned* cnt    = (unsigned*)d_ws;  // 384 u32 scratch counters
  unsigned* g_pred = cnt;
  unsigned* g_true = cnt + 128;
  unsigned* g_tp   = cnt + 256;

  mf1_zero<<<1, 384, 0, stream>>>(cnt);
  mf1_main<<<4096, 256, 0, stream>>>(logits, labels, g_pred, g_true, g_tp, N);
  mf1_final<<<1, 32, 0, stream>>>(g_pred, g_true, g_tp, (float*)d_out);
}